// EquiGroupSamplingIco_69037304315942
// MI455X (gfx1250) — compile-verified
//
#include <hip/hip_runtime.h>

typedef __attribute__((ext_vector_type(2))) float v2f;
typedef __attribute__((ext_vector_type(8))) float v8f;

__device__ __forceinline__ v8f zero8() {
    v8f z = {0.f,0.f,0.f,0.f,0.f,0.f,0.f,0.f};
    return z;
}

// D = A(16x4) * B(4x16) + C, all f32, one matrix per wave (wave32).
__device__ __forceinline__ v8f wmma4(v2f a, v2f b, v8f c) {
    return __builtin_amdgcn_wmma_f32_16x16x4_f32(
        /*neg_a=*/false, a, /*neg_b=*/false, b,
        /*c_mod=*/(short)0, c, /*reuse_a=*/false, /*reuse_b=*/false);
}

#define DIM455 455

// ---------------------------------------------------------------------------
// Kernel A: per-(b,t) fused chain:
//   h(16x10) -> relu(h * Din^T)(16x4096) -> *Doutw -> H2(16x455)
//   -> w2 equivariant mix (per l) -> ico projection -> out traj (64x60)
// ---------------------------------------------------------------------------
__global__ __launch_bounds__(256) void traj_kernel(
    const float* __restrict__ traj, const float* __restrict__ w1s,
    const float* __restrict__ w1v,  const float* __restrict__ Din,
    const float* __restrict__ Dout, const float* __restrict__ ico,
    const float* __restrict__ w20, const float* __restrict__ w21,
    const float* __restrict__ w22, const float* __restrict__ w23,
    const float* __restrict__ w24, const float* __restrict__ w25,
    const float* __restrict__ w26, float* __restrict__ out)
{
    constexpr int G = 4096, KC = 128;
    __shared__ float h_lds[16][12];     // 16x10 padded
    __shared__ float S[16][130];        // sigma chunk (16 x 128, padded stride)
    __shared__ float H2[16][456];       // 16 x 455 result of big GEMM
    __shared__ float sc[8][16][16];     // per-wave fragment relayout scratch

    const int bt   = blockIdx.x;        // 0..511  (b*16 + t)
    const int tid  = threadIdx.x;
    const int wave = tid >> 5;
    const int lane = tid & 31;
    const int lrow = lane & 15;         // M (A/C) or N (B) index within tile
    const int kb   = (lane >> 4) << 1;  // K sub-offset: 0 (lanes<16) / 2
    const int crow = (lane >> 4) << 3;  // C-fragment row offset: 0 / 8

    const float* t10 = traj + bt * 10;

    // ---- stage 1: h[c][i] = concat(equi_linear(s,w1s), equi_linear(v,w1v))
    if (tid < 160) {
        int c = tid / 10, i = tid % 10;
        float hv;
        if (i == 0) {
            hv = w1s[c] * t10[9];                       // fan_in = 1
        } else {
            int vv = (i - 1) / 3, mm = (i - 1) % 3;
            float a = 0.f;
            #pragma unroll
            for (int u = 0; u < 3; ++u)
                a += t10[u * 3 + mm] * w1v[c * 9 + u * 3 + vv];
            hv = a * 0.5773502691896258f;               // 1/sqrt(3)
        }
        h_lds[c][i] = hv;
    }
    __syncthreads();

    // ---- stages 2+3: H2 = relu(h * Din^T) * Doutw, chunked over G
    v8f acc0 = zero8(), acc1 = zero8(), acc2 = zero8(), acc3 = zero8();

    for (int g0 = 0; g0 < G; g0 += KC) {
        // stage 2 (VALU): sigma chunk into LDS
        for (int idx = tid; idx < 16 * KC; idx += 256) {
            int c = idx >> 7, gg = idx & (KC - 1);
            const float* dr = Din + (g0 + gg) * 10;
            float a = 0.f;
            #pragma unroll
            for (int i = 0; i < 10; ++i) a += h_lds[c][i] * dr[i];
            S[c][gg] = fmaxf(a, 0.f);
        }
        __syncthreads();

        // stage 3 (WMMA): unconditional 4 tiles/wave; tail tile uses
        // clamped addresses + value select (no exec-masked loads).
        for (int kk = 0; kk < KC; kk += 4) {
            v2f a;
            a.x = S[lrow][kk + kb];
            a.y = S[lrow][kk + kb + 1];
            int k = g0 + kk + kb;
            const float* d0 = Dout + k * DIM455;
            const float* d1 = Dout + (k + 1) * DIM455;

            int n0 = (wave + 0 * 8) * 16 + lrow;
            int n1 = (wave + 1 * 8) * 16 + lrow;
            int n2 = (wave + 2 * 8) * 16 + lrow;
            int n3 = (wave + 3 * 8) * 16 + lrow;
            int n3c = (n3 < DIM455) ? n3 : (DIM455 - 1);
            v2f b0, b1, b2, b3;
            b0.x = d0[n0];   b0.y = d1[n0];             // n0 <= 127 valid
            b1.x = d0[n1];   b1.y = d1[n1];             // n1 <= 255 valid
            b2.x = d0[n2];   b2.y = d1[n2];             // n2 <= 383 valid
            float t3x = d0[n3c], t3y = d1[n3c];         // unconditional, in-bounds
            b3.x = (n3 < DIM455) ? t3x : 0.f;
            b3.y = (n3 < DIM455) ? t3y : 0.f;

            acc0 = wmma4(a, b0, acc0);
            acc1 = wmma4(a, b1, acc1);
            acc2 = wmma4(a, b2, acc2);
            acc3 = wmma4(a, b3, acc3);
        }
        __syncthreads();
    }

    // store H2 fragments (C layout: vgpr j -> row j + crow, col = lane&15)
    {
        v8f* accs[4] = {&acc0, &acc1, &acc2, &acc3};
        #pragma unroll
        for (int tt = 0; tt < 4; ++tt) {
            int col = (wave + tt * 8) * 16 + lrow;
            if (col < DIM455) {
                #pragma unroll
                for (int j = 0; j < 8; ++j)
                    H2[j + crow][col] = (*accs[tt])[j];
            }
        }
    }
    __syncthreads();

    // ---- stages 5+6: per-l w2 mix (WMMA) fused with ico projection (WMMA)
    const int mtile = wave >> 1;   // g2 tile (4 tiles of 16 -> 64 channels)
    const int sub   = wave & 1;    // 2 waves share a tile, split (l,v) tasks

    v8f facc0 = zero8(), facc1 = zero8(), facc2 = zero8(), facc3 = zero8();

    #pragma unroll
    for (int l = 0; l <= 6; ++l) {
        const int d = 2 * l + 1;                // compile-time after unroll
        const int OFFT[7] = {0, 1, 10, 35, 84, 165, 286};
        const int o = OFFT[l];
        const int Kdim = 16 * d;
        const float scale = rsqrtf((float)(16 * d));
        const float* w2l = (l == 0) ? w20 : (l == 1) ? w21 : (l == 2) ? w22 :
                           (l == 3) ? w23 : (l == 4) ? w24 : (l == 5) ? w25 : w26;

        // tasks before level l number l^2 (l^2 ≡ l mod 2): parity split
        for (int v = (sub + l) & 1; v < d; v += 2) {
            // stage 5: out_l[g2, m] = sum_{f,u} H2[f][o+u*d+m] * w2l[f,g2,u,v]
            v8f dacc = zero8();
            const int m  = lrow;
            const int mc = (m < d) ? m : (d - 1);       // clamped column
            for (int k0 = 0; k0 < Kdim; k0 += 4) {
                int k  = k0 + kb;
                int f0 = k & 15,        u0 = k >> 4;
                int f1 = (k + 1) & 15,  u1 = (k + 1) >> 4;
                v2f a;
                a.x = w2l[((f0 * 64 + mtile * 16 + lrow) * d + u0) * d + v];
                a.y = w2l[((f1 * 64 + mtile * 16 + lrow) * d + u1) * d + v];
                float h0 = H2[f0][o + u0 * d + mc];
                float h1 = H2[f1][o + u1 * d + mc];
                v2f b;
                b.x = (m < d) ? h0 : 0.f;
                b.y = (m < d) ? h1 : 0.f;
                dacc = wmma4(a, b, dacc);
            }

            // relayout D-fragment -> A-matrix via per-wave scratch (scaled, zero-padded)
            #pragma unroll
            for (int j = 0; j < 8; ++j)
                sc[wave][j + crow][lrow] = (lrow < d) ? dacc[j] * scale : 0.f;

            // stage 6: facc[g2, r] += out_l[g2, m] * ico[r, o + v*d + m]
            #pragma unroll
            for (int k0 = 0; k0 < 16; k0 += 4) {
                int k   = k0 + kb;
                int kc0 = (k     < d) ? k       : (d - 1);
                int kc1 = (k + 1 < d) ? (k + 1) : (d - 1);
                v2f a;
                a.x = sc[wave][lrow][k];
                a.y = sc[wave][lrow][k + 1];
                const float* ib = ico + o + v * d;
                int r0 = 0 * 16 + lrow, r1 = 1 * 16 + lrow;
                int r2 = 2 * 16 + lrow, r3 = 3 * 16 + lrow;
                int r3c = (r3 < 60) ? r3 : 59;
                float t0x = ib[r0 * DIM455 + kc0], t0y = ib[r0 * DIM455 + kc1];
                float t1x = ib[r1 * DIM455 + kc0], t1y = ib[r1 * DIM455 + kc1];
                float t2x = ib[r2 * DIM455 + kc0], t2y = ib[r2 * DIM455 + kc1];
                float t3x = ib[r3c * DIM455 + kc0], t3y = ib[r3c * DIM455 + kc1];
                v2f b0, b1, b2, b3;
                b0.x = (k     < d) ? t0x : 0.f;
                b0.y = (k + 1 < d) ? t0y : 0.f;
                b1.x = (k     < d) ? t1x : 0.f;
                b1.y = (k + 1 < d) ? t1y : 0.f;
                b2.x = (k     < d) ? t2x : 0.f;
                b2.y = (k + 1 < d) ? t2y : 0.f;
                b3.x = (k     < d && r3 < 60) ? t3x : 0.f;
                b3.y = (k + 1 < d && r3 < 60) ? t3y : 0.f;
                facc0 = wmma4(a, b0, facc0);
                facc1 = wmma4(a, b1, facc1);
                facc2 = wmma4(a, b2, facc2);
                facc3 = wmma4(a, b3, facc3);
            }
        }
    }
    __syncthreads();

    // reduce the two waves per mtile through LDS (reuse H2 storage), write out
    float* red = &H2[0][0];                       // 64 x 64 floats
    {
        v8f* faccs[4] = {&facc0, &facc1, &facc2, &facc3};
        if (sub == 1) {
            #pragma unroll
            for (int nt = 0; nt < 4; ++nt)
                #pragma unroll
                for (int j = 0; j < 8; ++j)
                    red[(mtile * 16 + j + crow) * 64 + nt * 16 + lrow] = (*faccs[nt])[j];
        }
        __syncthreads();
        if (sub == 0) {
            const long xg_elems = 32L * 4 * 128 * 60;   // xg output precedes traj
            #pragma unroll
            for (int nt = 0; nt < 4; ++nt) {
                int r = nt * 16 + lrow;
                if (r < 60) {
                    #pragma unroll
                    for (int j = 0; j < 8; ++j) {
                        int g2 = mtile * 16 + j + crow;
                        float vsum = (*faccs[nt])[j] + red[g2 * 64 + r];
                        out[xg_elems + ((long)bt * 64 + g2) * 60 + r] = vsum;
                    }
                }
            }
        }
    }
}

// ---------------------------------------------------------------------------
// Kernel B: xg = x(16384 x 455) * ico^T(455 x 60)   (memory-bound WMMA GEMM)
// ---------------------------------------------------------------------------
__global__ __launch_bounds__(256) void xg_kernel(
    const float* __restrict__ x, const float* __restrict__ ico,
    float* __restrict__ out)
{
    const int tid  = threadIdx.x;
    const int wave = tid >> 5;
    const int lane = tid & 31;
    const int lrow = lane & 15;
    const int kb   = (lane >> 4) << 1;
    const int crow = (lane >> 4) << 3;
    const int mtile = wave >> 1;          // 4 M-tiles of 16 rows per block
    const int ng    = wave & 1;           // N-tile group {0,1} or {2,3}

    const long rowA = (long)blockIdx.x * 64 + mtile * 16 + lrow;
    const float* xr = x + rowA * DIM455;

    v8f acc0 = zero8(), acc1 = zero8();
    const int r0  = (ng * 2 + 0) * 16 + lrow;     // <= 47, always valid
    const int r1  = (ng * 2 + 1) * 16 + lrow;
    const int r1c = (r1 < 60) ? r1 : 59;

    for (int k0 = 0; k0 < 456; k0 += 4) {
        int k   = k0 + kb;
        int kc0 = (k     < DIM455) ? k       : (DIM455 - 1);
        int kc1 = (k + 1 < DIM455) ? (k + 1) : (DIM455 - 1);
        float ax = xr[kc0], ay = xr[kc1];
        float b0x = ico[r0  * DIM455 + kc0], b0y = ico[r0  * DIM455 + kc1];
        float b1x = ico[r1c * DIM455 + kc0], b1y = ico[r1c * DIM455 + kc1];
        v2f a, b0, b1;
        a.x  = (k     < DIM455) ? ax  : 0.f;
        a.y  = (k + 1 < DIM455) ? ay  : 0.f;
        b0.x = (k     < DIM455) ? b0x : 0.f;
        b0.y = (k + 1 < DIM455) ? b0y : 0.f;
        b1.x = (k     < DIM455 && r1 < 60) ? b1x : 0.f;
        b1.y = (k + 1 < DIM455 && r1 < 60) ? b1y : 0.f;
        acc0 = wmma4(a, b0, acc0);
        acc1 = wmma4(a, b1, acc1);
    }

    #pragma unroll
    for (int s = 0; s < 2; ++s) {
        int r = (ng * 2 + s) * 16 + lrow;
        if (r < 60) {
            #pragma unroll
            for (int j = 0; j < 8; ++j) {
                long row = (long)blockIdx.x * 64 + mtile * 16 + j + crow;
                out[row * 60 + r] = (s == 0) ? acc0[j] : acc1[j];
            }
        }
    }
}

// ---------------------------------------------------------------------------
extern "C" void kernel_launch(void* const* d_in, const int* in_sizes, int n_in,
                              void* d_out, int out_size, void* d_ws, size_t ws_size,
                              hipStream_t stream) {
    // setup_inputs() dict order:
    // 0:x 1:trajectory 2:w1s 3:w1v 4:D_in 5:D_outw 6:ico_wigners 7..13:w2_0..w2_6
    const float* x    = (const float*)d_in[0];
    const float* traj = (const float*)d_in[1];
    const float* w1s  = (const float*)d_in[2];
    const float* w1v  = (const float*)d_in[3];
    const float* Din  = (const float*)d_in[4];
    const float* Dout = (const float*)d_in[5];
    const float* ico  = (const float*)d_in[6];
    float* out = (float*)d_out;

    // xg: 32*4*128 = 16384 rows, 64 rows per block
    xg_kernel<<<dim3(16384 / 64), dim3(256), 0, stream>>>(x, ico, out);

    // traj path: one block per (b,t)
    traj_kernel<<<dim3(512), dim3(256), 0, stream>>>(
        traj, w1s, w1v, Din, Dout, ico,
        (const float*)d_in[7],  (const float*)d_in[8],  (const float*)d_in[9],
        (const float*)d_in[10], (const float*)d_in[11], (const float*)d_in[12],
        (const float*)d_in[13], out);
}